// RPNPostProcessor_52999896433082
// MI455X (gfx1250) — compile-verified
//
#include <hip/hip_runtime.h>
#include <stdint.h>

#define NB 16
#define AA 15
#define HH 128
#define WW 128
#define NUM (AA*HH*WW)       // 245760
#define PRE 2000
#define PREP 2048
#define POST 1000
#define NMS_THR 0.7f
#define MIN_SZ 4.0f
#define NEGV (-1000000000.0f)
#define XCLIP 4.135166556742356f   // log(1000/16)

typedef unsigned int u32;
typedef unsigned long long u64;

typedef __attribute__((ext_vector_type(2))) float v2f;
typedef __attribute__((ext_vector_type(8))) float v8f;

// ---- orderable key transform: larger float -> larger u32 key ----
__device__ __forceinline__ u32 fkey(float x) {
  u32 b = __float_as_uint(x);
  return b ^ ((b & 0x80000000u) ? 0xFFFFFFFFu : 0x80000000u);
}
__device__ __forceinline__ float kinv(u32 u) {
  u32 b = u ^ ((u & 0x80000000u) ? 0x80000000u : 0xFFFFFFFFu);
  return __uint_as_float(b);
}

// =====================================================================
// Kernel 1: exact top-2000 per image via 4-round radix select.
// 256-bin inclusive prefix scan done with V_WMMA_F32_16X16X4_F32:
//   P = H*U + Lstrict*(H*Ones)   (16x16 matrices, K sliced by 4)
// =====================================================================
__global__ __launch_bounds__(1024) void k_select(
    const float* __restrict__ objL, const float* __restrict__ objR,
    u32* __restrict__ keys, u32* __restrict__ idxs)
{
  const int img = blockIdx.x;
  const int side = img >> 4, n = img & 15;
  const float* obj = (side ? objR : objL) + (size_t)n * NUM;
  const int tid = threadIdx.x;

  __shared__ u32   hist[256];
  __shared__ float prefixf[256];
  __shared__ float rowsum[16];
  __shared__ u32   tiebuf[PREP];
  __shared__ int   s_bin;
  __shared__ float s_above;
  __shared__ u32   s_cgt, s_ctie;

  u32 pfx = 0;
  int kwant = PRE;

  for (int r = 0; r < 4; ++r) {
    for (int b = tid; b < 256; b += 1024) hist[b] = 0u;
    __syncthreads();
    const int sh = 24 - 8 * r;
    for (int m = tid; m < NUM; m += 1024) {
      u32 u = fkey(obj[m]);
      bool part = (r == 0) || ((u >> (sh + 8)) == pfx);
      if (part) atomicAdd(&hist[(u >> sh) & 255u], 1u);
    }
    __syncthreads();
    if (tid < 16) {
      float s = 0.f;
      for (int c = 0; c < 16; ++c) s += (float)hist[tid * 16 + c];
      rowsum[tid] = s;
    }
    __syncthreads();
#if __has_builtin(__builtin_amdgcn_wmma_f32_16x16x4_f32)
    if (tid < 32) {                       // wave 0, EXEC all ones
      const int lane = tid, lm = lane & 15;
      const bool hi = lane >= 16;
      v8f acc = {};
      // P1 = M x U (upper-triangular ones): within-row inclusive scan
      #pragma unroll
      for (int s = 0; s < 4; ++s) {
        const int k0 = 4 * s + (hi ? 2 : 0);
        const int rx = 4 * s + (hi ? 1 : 0);
        const int ry = 4 * s + (hi ? 3 : 2);
        v2f a, b;
        a.x = (float)hist[lm * 16 + k0];
        a.y = (float)hist[lm * 16 + k0 + 1];
        b.x = (rx <= lm) ? 1.f : 0.f;
        b.y = (ry <= lm) ? 1.f : 0.f;
        acc = __builtin_amdgcn_wmma_f32_16x16x4_f32(
            false, a, false, b, (short)0, acc, false, false);
      }
      // P2 += Lstrict x T,  T[k][c] = rowsum(k): add totals of prior rows
      #pragma unroll
      for (int s = 0; s < 4; ++s) {
        const int k0 = 4 * s + (hi ? 2 : 0);
        const int rx = 4 * s + (hi ? 1 : 0);
        const int ry = 4 * s + (hi ? 3 : 2);
        v2f a, b;
        a.x = (k0     < lm) ? 1.f : 0.f;
        a.y = (k0 + 1 < lm) ? 1.f : 0.f;
        b.x = rowsum[rx];
        b.y = rowsum[ry];
        acc = __builtin_amdgcn_wmma_f32_16x16x4_f32(
            false, a, false, b, (short)0, acc, false, false);
      }
      #pragma unroll
      for (int g = 0; g < 8; ++g) {
        int row = (hi ? 8 : 0) + g;       // C/D layout: VGPR g -> M=g / M=8+g
        prefixf[row * 16 + lm] = acc[g];
      }
    }
#else
    if (tid == 0) { float s = 0.f; for (int b = 0; b < 256; ++b) { s += (float)hist[b]; prefixf[b] = s; } }
#endif
    __syncthreads();
    if (tid < 256) {
      float total = prefixf[255];
      float above = total - prefixf[tid];     // count strictly above bin tid
      float h     = (float)hist[tid];
      float kw    = (float)kwant;
      if (above < kw && above + h >= kw) { s_bin = tid; s_above = above; }
    }
    __syncthreads();
    pfx = (pfx << 8) | (u32)s_bin;
    kwant -= (int)s_above;
    __syncthreads();
  }

  // ---- gather: all keys > T, plus `kwant` smallest-index ties == T ----
  if (tid == 0) { s_cgt = 0u; s_ctie = 0u; }
  __syncthreads();
  const u32 T = pfx;
  const int need = kwant;
  for (int m = tid; m < NUM; m += 1024) {
    u32 u = fkey(obj[m]);
    if (u >= T) {
      int a  = m / (HH * WW);
      int hw = m - a * (HH * WW);
      u32 i  = (u32)(hw * AA + a);           // flat (h,w,a) index of reference
      if (u > T) {
        u32 p = atomicAdd(&s_cgt, 1u);
        keys[(size_t)img * PREP + p] = u;
        idxs[(size_t)img * PREP + p] = i;
      } else {
        u32 q = atomicAdd(&s_ctie, 1u);
        if (q < PREP) tiebuf[q] = i;
      }
    }
  }
  __syncthreads();
  const u32 ngt  = s_cgt;
  const u32 ntie = (s_ctie < (u32)PREP) ? s_ctie : (u32)PREP;
  for (int q = tid; q < PREP; q += 1024) if ((u32)q >= ntie) tiebuf[q] = 0xFFFFFFFFu;
  __syncthreads();
  // bitonic ascending sort of tie indices (deterministic tie-breaking)
  for (int k = 2; k <= PREP; k <<= 1) {
    for (int j = k >> 1; j > 0; j >>= 1) {
      for (int e = tid; e < PREP; e += 1024) {
        int p = e ^ j;
        if (p > e) {
          u32 x = tiebuf[e], y = tiebuf[p];
          bool up = ((e & k) == 0);
          if ((x > y) == up) { tiebuf[e] = y; tiebuf[p] = x; }
        }
      }
      __syncthreads();
    }
  }
  for (int t = tid; t < need; t += 1024) {
    keys[(size_t)img * PREP + ngt + t] = T;
    idxs[(size_t)img * PREP + ngt + t] = tiebuf[t];
  }
  for (int p = PRE + tid; p < PREP; p += 1024) {   // pad slots
    keys[(size_t)img * PREP + p] = 0u;
    idxs[(size_t)img * PREP + p] = 0xFFFFFFFFu;
  }
}

// =====================================================================
// Kernel 2: sort 2048 (key desc, idx asc), gather, decode, clip, valid
// =====================================================================
__global__ __launch_bounds__(1024) void k_decode(
    const float* __restrict__ ancL, const float* __restrict__ ancR,
    const float* __restrict__ brL,  const float* __restrict__ brR,
    const u32* __restrict__ keys, const u32* __restrict__ idxs,
    float* __restrict__ boxes4, float* __restrict__ scores,
    u32* __restrict__ keepinit)
{
  const int img = blockIdx.x;
  const int side = img >> 4, n = img & 15;
  const float* anc = side ? ancR : ancL;
  const float* br  = side ? brR  : brL;
  const int tid = threadIdx.x;
  __shared__ u64 arr[PREP];
  __shared__ u32 kw[64];

  for (int e = tid; e < PREP; e += 1024) {
    u32 k = keys[(size_t)img * PREP + e];
    u32 i = idxs[(size_t)img * PREP + e];
    arr[e] = ((u64)k << 32) | (u32)(~i);   // desc key, asc idx
  }
  if (tid < 64) kw[tid] = 0u;
  __syncthreads();
  for (int k = 2; k <= PREP; k <<= 1) {
    for (int j = k >> 1; j > 0; j >>= 1) {
      for (int e = tid; e < PREP; e += 1024) {
        int p = e ^ j;
        if (p > e) {
          u64 x = arr[e], y = arr[p];
          bool up = ((e & k) == 0);
          if ((x < y) == up) { arr[e] = y; arr[p] = x; }   // overall descending
        }
      }
      __syncthreads();
    }
  }
  for (int r = tid; r < PREP; r += 1024) {
    size_t ob = (size_t)img * PREP * 4 + (size_t)r * 4;
    if (r < PRE) {
      u64 c = arr[r];
      u32 key = (u32)(c >> 32);
      u32 i   = ~((u32)c);
      float logit = kinv(key);
      float sc = 1.0f / (1.0f + __expf(-logit));
      const float4 av = ((const float4*)anc)[(size_t)n * NUM + i];
      int a  = (int)(i % AA);
      int hw = (int)(i / AA);
      int w  = hw & (WW - 1);
      int h  = hw >> 7;
      size_t bo = ((((size_t)n * AA + a) * 4) * HH + h) * WW + w;
      const size_t cs = (size_t)HH * WW;
      float dx = br[bo], dy = br[bo + cs];
      float dw = br[bo + 2 * cs], dh = br[bo + 3 * cs];
      float wa = av.z - av.x + 1.0f;
      float ha = av.w - av.y + 1.0f;
      float cx = av.x + 0.5f * wa;
      float cy = av.y + 0.5f * ha;
      dw = fminf(dw, XCLIP); dh = fminf(dh, XCLIP);
      float pcx = dx * wa + cx, pcy = dy * ha + cy;
      float pw = __expf(dw) * wa, ph = __expf(dh) * ha;
      float x1 = pcx - 0.5f * pw, y1 = pcy - 0.5f * ph;
      float x2 = pcx + 0.5f * pw - 1.0f, y2 = pcy + 0.5f * ph - 1.0f;
      x1 = fminf(fmaxf(x1, 0.f), 1023.f);
      y1 = fminf(fmaxf(y1, 0.f), 1023.f);
      x2 = fminf(fmaxf(x2, 0.f), 1023.f);
      y2 = fminf(fmaxf(y2, 0.f), 1023.f);
      boxes4[ob] = x1; boxes4[ob + 1] = y1; boxes4[ob + 2] = x2; boxes4[ob + 3] = y2;
      scores[(size_t)img * PREP + r] = sc;
      bool valid = ((x2 - x1 + 1.f) >= MIN_SZ) && ((y2 - y1 + 1.f) >= MIN_SZ);
      if (valid) atomicOr(&kw[r >> 5], 1u << (r & 31));
    } else {
      boxes4[ob] = 0.f; boxes4[ob + 1] = 0.f; boxes4[ob + 2] = 0.f; boxes4[ob + 3] = 0.f;
      scores[(size_t)img * PREP + r] = NEGV;
    }
  }
  __syncthreads();
  if (tid < 64) keepinit[img * 64 + tid] = kw[tid];
}

// =====================================================================
// Kernel 3a: suppression bitmask (parallel over whole grid).
// Box tile (32KB) staged to LDS via the Tensor Data Mover when available.
// =====================================================================
#if __has_builtin(__builtin_amdgcn_tensor_load_to_lds) && __has_builtin(__builtin_amdgcn_s_wait_tensorcnt)
#define HAVE_TDM 1
typedef unsigned int u32x4 __attribute__((ext_vector_type(4)));
typedef int i32x8 __attribute__((ext_vector_type(8)));
typedef int i32x4 __attribute__((ext_vector_type(4)));
#else
#define HAVE_TDM 0
#endif

__global__ __launch_bounds__(256) void k_mask(
    const float* __restrict__ boxes4, u32* __restrict__ mask)
{
  const int img   = blockIdx.x / 125;
  const int chunk = blockIdx.x % 125;   // 16 rows per chunk, 125*16 = 2000
  const int tid   = threadIdx.x;
  __shared__ float4 sbox[PREP];
  const float* bp = boxes4 + (size_t)img * PREP * 4;
#if HAVE_TDM
  if (tid < 32) {  // wave 0 issues one TDM op (EXEC ignored by TDM)
    unsigned long long ga = (unsigned long long)(uintptr_t)bp;
    unsigned int lds = (unsigned int)(uintptr_t)(&sbox[0]);
    u32x4 g0; i32x8 g1; i32x4 g2 = {0, 0, 0, 0}, g3 = {0, 0, 0, 0};
    i32x8 g4 = {0, 0, 0, 0, 0, 0, 0, 0};             // extension group (unused)
    g0[0] = 1u;                                      // count=1 (valid user D#)
    g0[1] = lds;                                     // lds_addr
    g0[2] = (unsigned int)(ga & 0xFFFFFFFFull);      // global_addr[31:0]
    g0[3] = (unsigned int)((ga >> 32) & 0x01FFFFFFull) | (2u << 30); // addr[56:32]|type=2
    g1[0] = (int)(2u << 16);                         // data_size = 4B
    g1[1] = (int)((8192u & 0xFFFFu) << 16);          // tensor_dim0[15:0] = 8192
    g1[2] = (int)((8192u >> 16) | (1u << 16));       // tensor_dim0[31:16], tensor_dim1=1
    g1[3] = (int)(0x2000u << 16);                    // tile_dim0 = 8192
    g1[4] = 1;                                       // tile_dim1 = 1, tile_dim2 = 0
    g1[5] = 8192;                                    // tensor_dim0_stride[31:0]
    g1[6] = (int)((8192u & 0xFFFFu) << 16);          // tensor_dim1_stride[15:0]
    g1[7] = 0;
    __builtin_amdgcn_tensor_load_to_lds(g0, g1, g2, g3, g4, 0);
    __builtin_amdgcn_s_wait_tensorcnt(0);
  }
  __syncthreads();
#else
  for (int e = tid; e < PREP; e += 256) sbox[e] = ((const float4*)bp)[e];
  __syncthreads();
#endif
  for (int q = tid; q < 16 * 64; q += 256) {
    int i  = chunk * 16 + (q >> 6);
    int wc = q & 63;
    float4 bi = sbox[i];
    float ai = (bi.z - bi.x + 1.f) * (bi.w - bi.y + 1.f);
    u32 mword = 0u;
    #pragma unroll 4
    for (int b = 0; b < 32; ++b) {
      int j = wc * 32 + b;
      if (j > i && j < PRE) {
        float4 bj = sbox[j];
        float aj = (bj.z - bj.x + 1.f) * (bj.w - bj.y + 1.f);
        float xx1 = fmaxf(bi.x, bj.x), yy1 = fmaxf(bi.y, bj.y);
        float xx2 = fminf(bi.z, bj.z), yy2 = fminf(bi.w, bj.w);
        float inter = fmaxf(xx2 - xx1 + 1.f, 0.f) * fmaxf(yy2 - yy1 + 1.f, 0.f);
        float iou = inter / (ai + aj - inter);
        if (iou > NMS_THR) mword |= (1u << b);
      }
    }
    mask[((size_t)img * PREP + i) * 64 + wc] = mword;
  }
}

// Kernel 3b: sequential keep propagation, one wave per image (mask in L2)
__global__ __launch_bounds__(32) void k_scan(
    const u32* __restrict__ keepinit, const u32* __restrict__ mask,
    u32* __restrict__ keepfin)
{
  const int img  = blockIdx.x;
  const int lane = threadIdx.x;
  __shared__ u32 kw[64];
  kw[lane]      = keepinit[img * 64 + lane];
  kw[lane + 32] = keepinit[img * 64 + lane + 32];
  __syncthreads();
  const u64* mbase = (const u64*)(mask + (size_t)img * PREP * 64);
  for (int i = 0; i < PRE; ++i) {
    u64 mv = mbase[(size_t)i * 32 + lane];                 // 64 words / row
    if (i + 8 < PRE) __builtin_prefetch(&mbase[(size_t)(i + 8) * 32 + lane], 0, 0);
    bool ki = (kw[i >> 5] >> (i & 31)) & 1u;               // uniform broadcast
    if (ki) {
      kw[2 * lane]     &= ~((u32)mv);
      kw[2 * lane + 1] &= ~((u32)(mv >> 32));
    }
  }
  __syncthreads();
  keepfin[img * 64 + lane]      = kw[lane];
  keepfin[img * 64 + lane + 32] = kw[lane + 32];
}

// Fallback: in-LDS sequential NMS if workspace can't hold the bitmask
__global__ __launch_bounds__(256) void k_nms_seq(
    const float* __restrict__ boxes4, const u32* __restrict__ keepinit,
    u32* __restrict__ keepfin)
{
  const int img = blockIdx.x;
  const int tid = threadIdx.x;
  __shared__ float4 sbox[PREP];
  __shared__ u32 kw[64];
  const float* bp = boxes4 + (size_t)img * PREP * 4;
  for (int e = tid; e < PREP; e += 256) sbox[e] = ((const float4*)bp)[e];
  if (tid < 64) kw[tid] = keepinit[img * 64 + tid];
  __syncthreads();
  for (int i = 0; i < PRE; ++i) {
    bool ki = (kw[i >> 5] >> (i & 31)) & 1u;
    if (ki) {
      float4 bi = sbox[i];
      float ai = (bi.z - bi.x + 1.f) * (bi.w - bi.y + 1.f);
      for (int j = i + 1 + tid; j < PRE; j += 256) {
        float4 bj = sbox[j];
        float aj = (bj.z - bj.x + 1.f) * (bj.w - bj.y + 1.f);
        float xx1 = fmaxf(bi.x, bj.x), yy1 = fmaxf(bi.y, bj.y);
        float xx2 = fminf(bi.z, bj.z), yy2 = fminf(bi.w, bj.w);
        float inter = fmaxf(xx2 - xx1 + 1.f, 0.f) * fmaxf(yy2 - yy1 + 1.f, 0.f);
        float iou = inter / (ai + aj - inter);
        if (iou > NMS_THR) atomicAnd(&kw[j >> 5], ~(1u << (j & 31)));
      }
    }
    __syncthreads();
  }
  if (tid < 64) keepfin[img * 64 + tid] = kw[tid];
}

// =====================================================================
// Kernel 4: compaction — kept (score) first, suppressed (NEG) after
// =====================================================================
__global__ __launch_bounds__(256) void k_final(
    const float* __restrict__ boxes4, const float* __restrict__ scores,
    const u32* __restrict__ keepfin, float* __restrict__ out)
{
  const int img = blockIdx.x;
  const int tid = threadIdx.x;
  __shared__ u32 kwords[64];
  __shared__ int part[256];
  if (tid < 64) kwords[tid] = keepfin[img * 64 + tid];
  __syncthreads();
  int cnt = 0;
  bool kept[8];
  #pragma unroll
  for (int s = 0; s < 8; ++s) {
    int r = tid * 8 + s;
    bool k = (r < PRE) && ((kwords[r >> 5] >> (r & 31)) & 1u);
    kept[s] = k;
    cnt += k ? 1 : 0;
  }
  part[tid] = cnt;
  __syncthreads();
  for (int d = 1; d < 256; d <<= 1) {
    int add = (tid >= d) ? part[tid - d] : 0;
    __syncthreads();
    part[tid] += add;
    __syncthreads();
  }
  const int totk = part[255];
  int run = part[tid] - cnt;               // kept before this thread's range
  #pragma unroll
  for (int s = 0; s < 8; ++s) {
    int r = tid * 8 + s;
    if (r >= PRE) break;
    int pos; float sc;
    if (kept[s]) { pos = run; sc = scores[(size_t)img * PREP + r]; run++; }
    else         { pos = totk + (r - run); sc = NEGV; }
    if (pos < POST) {
      size_t ob = (size_t)img * PREP * 4 + (size_t)r * 4;
      size_t oo = ((size_t)img * POST + pos) * 5;
      out[oo]     = boxes4[ob];
      out[oo + 1] = boxes4[ob + 1];
      out[oo + 2] = boxes4[ob + 2];
      out[oo + 3] = boxes4[ob + 3];
      out[oo + 4] = sc;
    }
  }
}

extern "C" void kernel_launch(void* const* d_in, const int* in_sizes, int n_in,
                              void* d_out, int out_size, void* d_ws, size_t ws_size,
                              hipStream_t stream) {
  const float* ancL = (const float*)d_in[0];
  const float* ancR = (const float*)d_in[1];
  const float* objL = (const float*)d_in[2];
  const float* objR = (const float*)d_in[3];
  const float* brL  = (const float*)d_in[4];
  const float* brR  = (const float*)d_in[5];
  float* out = (float*)d_out;

  u32* ws = (u32*)d_ws;
  const size_t NIMG = 32;
  size_t off = 0;
  u32*   keys     = ws + off;              off += NIMG * PREP;
  u32*   idxs     = ws + off;              off += NIMG * PREP;
  float* boxes4   = (float*)(ws + off);    off += NIMG * PREP * 4;
  float* scores   = (float*)(ws + off);    off += NIMG * PREP;
  u32*   keepinit = ws + off;              off += NIMG * 64;
  u32*   keepfin  = ws + off;              off += NIMG * 64;
  u32*   mask     = ws + off;
  const size_t mask_words = NIMG * (size_t)PREP * 64;   // 16 MB
  const bool use_mask = ws_size >= (off + mask_words) * sizeof(u32);

  k_select<<<32, 1024, 0, stream>>>(objL, objR, keys, idxs);
  k_decode<<<32, 1024, 0, stream>>>(ancL, ancR, brL, brR, keys, idxs,
                                    boxes4, scores, keepinit);
  if (use_mask) {
    k_mask<<<32 * 125, 256, 0, stream>>>(boxes4, mask);
    k_scan<<<32, 32, 0, stream>>>(keepinit, mask, keepfin);
  } else {
    k_nms_seq<<<32, 256, 0, stream>>>(boxes4, keepinit, keepfin);
  }
  k_final<<<32, 256, 0, stream>>>(boxes4, scores, keepfin, out);

  (void)in_sizes; (void)n_in; (void)out_size;
}